// GATModel_20461224198262
// MI455X (gfx1250) — compile-verified
//
#include <hip/hip_runtime.h>
#include <cmath>

#define N_NODES 50000
#define N_EDGES 400000
#define N_GRAPH 64
#define IN_CH   128
#define HID     64
#define NHEAD   4
#define CPROJ   256   // NHEAD * HID
#define NEG_SLOPE 0.2f

typedef float v2f __attribute__((ext_vector_type(2)));
typedef float v8f __attribute__((ext_vector_type(8)));

// ---------------------------------------------------------------- fill
__global__ void fill_u32(unsigned int* __restrict__ p, unsigned int v, long n) {
    long i = (long)blockIdx.x * blockDim.x + threadIdx.x;
    if (i < n) p[i] = v;
}

// ------------------------------------------------- edge attr embedding
__global__ void edge_embed(const float* __restrict__ ea,
                           const float* __restrict__ W_et, const float* __restrict__ b_et,
                           const float* __restrict__ W_uw, const float* __restrict__ b_uw,
                           const float* __restrict__ W_ua, const float* __restrict__ b_ua,
                           float* __restrict__ eap) {
    int e = blockIdx.x * blockDim.x + threadIdx.x;
    if (e >= N_EDGES) return;
    float et = ea[2 * e + 0];
    float uw = ea[2 * e + 1];
    float ua = 1.f / (1.f + __expf(-(uw * W_ua[0] + b_ua[0])));
    eap[4 * e + 0] = et * W_et[0] + b_et[0];
    eap[4 * e + 1] = et * W_et[1] + b_et[1];
    eap[4 * e + 2] = (uw * W_uw[0] + b_uw[0]) * ua;
    eap[4 * e + 3] = (uw * W_uw[1] + b_uw[1]) * ua;
}

// -------------------------------------- fp32 WMMA GEMM: Y = X(MxK) * W(Kx256) + b
// one wave per 16-row tile; 4 column tiles (16x64 out) per wave, A reused 4x.
__global__ __launch_bounds__(32) void gemm_wmma_f32(
        const float* __restrict__ X, const float* __restrict__ W,
        const float* __restrict__ bias, float* __restrict__ Y, int K) {
    const int lane = threadIdx.x;      // 0..31, wave32
    const int half = lane >> 4;        // 0: K=k,k+1   1: K=k+2,k+3
    const int l15  = lane & 15;
    const int row  = blockIdx.x * 16 + l15;       // A row (M)
    const int colBase = blockIdx.y * 64;          // 4 x 16-col tiles
    const float* Xr = X + (long)row * K;

    v8f acc[4] = {};
    for (int k = 0; k < K; k += 4) {
        const int ka = k + half * 2;
        v2f a;
        a.x = Xr[ka];
        a.y = Xr[ka + 1];
        #pragma unroll
        for (int t = 0; t < 4; ++t) {
            const int col = colBase + t * 16 + l15;
            v2f b;
            b.x = W[(long)ka * CPROJ + col];
            b.y = W[(long)(ka + 1) * CPROJ + col];
            acc[t] = __builtin_amdgcn_wmma_f32_16x16x4_f32(
                         false, a, false, b, (short)0, acc[t], false, false);
        }
    }
    // C/D layout: VGPR r -> M = r (lanes 0-15) / r+8 (lanes 16-31), N = l15
    #pragma unroll
    for (int t = 0; t < 4; ++t) {
        const int col = colBase + t * 16 + l15;
        const float bv = bias[col];
        #pragma unroll
        for (int r = 0; r < 8; ++r) {
            const int orow = blockIdx.x * 16 + r + half * 8;
            Y[(long)orow * CPROJ + col] = acc[t][r] + bv;
        }
    }
}

// -------------------------- attention scores: one wave32 per (edge, head)
__global__ __launch_bounds__(256) void edge_scores(
        const float* __restrict__ xl, const float* __restrict__ xr,
        const float* __restrict__ eap, const float* __restrict__ We,
        const float* __restrict__ att,
        const int* __restrict__ src, const int* __restrict__ dst,
        float* __restrict__ score, float* __restrict__ smax) {
    const int gw   = (int)(((long)blockIdx.x * blockDim.x + threadIdx.x) >> 5);
    const int lane = threadIdx.x & 31;
    if (gw >= N_EDGES * NHEAD) return;
    const int e = gw >> 2, h = gw & 3;
    const int s = src[e], d = dst[e];
    const float e0 = eap[4 * e + 0], e1 = eap[4 * e + 1];
    const float e2 = eap[4 * e + 2], e3 = eap[4 * e + 3];
    float acc = 0.f;
    #pragma unroll
    for (int i = 0; i < 2; ++i) {
        const int c = h * HID + lane + i * 32;   // contiguous 64-ch stretch
        const float ee = e0 * We[c] + e1 * We[CPROJ + c] +
                         e2 * We[2 * CPROJ + c] + e3 * We[3 * CPROJ + c];
        float m = xl[(long)s * CPROJ + c] + xr[(long)d * CPROJ + c] + ee;
        m = (m > 0.f) ? m : NEG_SLOPE * m;
        acc += m * att[c];
    }
    #pragma unroll
    for (int o = 16; o > 0; o >>= 1) acc += __shfl_xor(acc, o, 32);
    if (lane == 0) {
        score[gw] = acc;
        float* addr = &smax[(long)d * NHEAD + h];
        if (acc >= 0.f) atomicMax((int*)addr, __float_as_int(acc));
        else            atomicMin((unsigned int*)addr, (unsigned int)__float_as_int(acc));
    }
}

// ------------------------------- exp + segment-sum denominator (score in place)
__global__ void edge_expsum(const float* score, const int* __restrict__ dst,
                            const float* __restrict__ smax,
                            float* exbuf, float* __restrict__ denom) {
    const int i = blockIdx.x * blockDim.x + threadIdx.x;
    if (i >= N_EDGES * NHEAD) return;
    const int e = i >> 2, h = i & 3;
    const int d = dst[e];
    const float ex = __expf(score[i] - smax[(long)d * NHEAD + h]);
    exbuf[i] = ex;
    atomicAdd(&denom[(long)d * NHEAD + h], ex);
}

// ------------------------- weighted scatter: one wave32 per (edge, head)
__global__ __launch_bounds__(256) void edge_accum(
        const float* __restrict__ xl,
        const int* __restrict__ src, const int* __restrict__ dst,
        const float* __restrict__ exbuf, const float* __restrict__ denom,
        float* __restrict__ agg) {
    const int gw   = (int)(((long)blockIdx.x * blockDim.x + threadIdx.x) >> 5);
    const int lane = threadIdx.x & 31;
    if (gw >= N_EDGES * NHEAD) return;
    const int e = gw >> 2, h = gw & 3;
    const int s = src[e], d = dst[e];
    const float alpha = exbuf[gw] / (denom[(long)d * NHEAD + h] + 1e-16f);
    #pragma unroll
    for (int i = 0; i < 2; ++i) {
        const int c = h * HID + lane + i * 32;
        atomicAdd(&agg[(long)d * CPROJ + c], xl[(long)s * CPROJ + c] * alpha);
    }
}

// ------------------------------------- head-mean + bias + relu -> N x 64
__global__ void node_finalize(const float* __restrict__ agg,
                              const float* __restrict__ bias,
                              float* __restrict__ out) {
    const int i = blockIdx.x * blockDim.x + threadIdx.x;
    if (i >= N_NODES * HID) return;
    const int n = i >> 6, c = i & 63;
    const float* a = agg + (long)n * CPROJ + c;
    const float v = 0.25f * (a[0] + a[64] + a[128] + a[192]) + bias[c];
    out[i] = fmaxf(v, 0.f);
}

// ----------------------------------------------- graph mean-pool scatter
__global__ void pool_scatter(const float* __restrict__ h, const int* __restrict__ batch,
                             float* __restrict__ pool, float* __restrict__ cnt) {
    const int i = blockIdx.x * blockDim.x + threadIdx.x;
    if (i >= N_NODES * HID) return;
    const int n = i >> 6, c = i & 63;
    const int g = batch[n];
    atomicAdd(&pool[g * HID + c], h[i]);
    if (c == 0) atomicAdd(&cnt[g], 1.f);
}

// --------------------------------------- per-graph MLP head (64 blocks x 64 thr)
__global__ __launch_bounds__(64) void graph_head(
        const float* __restrict__ pool, const float* __restrict__ cnt,
        const float* __restrict__ Wfc, const float* __restrict__ bfc,
        const float* __restrict__ Wres, const float* __restrict__ bres,
        const float* __restrict__ Wtime, const float* __restrict__ btime,
        float* __restrict__ out) {
    __shared__ float g[HID];
    const int gi = blockIdx.x, j = threadIdx.x;
    const float c = fmaxf(cnt[gi], 1.f);
    float acc = bfc[j];
    for (int k = 0; k < HID; ++k) acc += (pool[gi * HID + k] / c) * Wfc[k * HID + j];
    g[j] = fmaxf(acc, 0.f);
    __syncthreads();
    if (j < 2) {
        const float* Wv = j ? Wtime : Wres;
        float r = j ? btime[0] : bres[0];
        for (int k = 0; k < HID; ++k) r += g[k] * Wv[k];
        out[gi * 2 + j] = r;
    }
}

// =====================================================================
extern "C" void kernel_launch(void* const* d_in, const int* in_sizes, int n_in,
                              void* d_out, int out_size, void* d_ws, size_t ws_size,
                              hipStream_t stream) {
    (void)in_sizes; (void)n_in; (void)out_size; (void)ws_size;
    const float* x     = (const float*)d_in[0];
    const float* ea    = (const float*)d_in[1];
    const int*   ei    = (const int*)  d_in[2];
    const int*   batch = (const int*)  d_in[3];
    const float* W_et  = (const float*)d_in[4];
    const float* b_et  = (const float*)d_in[5];
    const float* W_uw  = (const float*)d_in[6];
    const float* b_uw  = (const float*)d_in[7];
    const float* W_ua  = (const float*)d_in[8];
    const float* b_ua  = (const float*)d_in[9];
    const float* Wl0   = (const float*)d_in[10];
    const float* bl0   = (const float*)d_in[11];
    const float* Wr0   = (const float*)d_in[12];
    const float* br0   = (const float*)d_in[13];
    const float* att0  = (const float*)d_in[14];
    const float* We0   = (const float*)d_in[15];
    const float* bias0 = (const float*)d_in[16];
    const float* Wl1   = (const float*)d_in[17];
    const float* bl1   = (const float*)d_in[18];
    const float* Wr1   = (const float*)d_in[19];
    const float* br1   = (const float*)d_in[20];
    const float* att1  = (const float*)d_in[21];
    const float* We1   = (const float*)d_in[22];
    const float* bias1 = (const float*)d_in[23];
    const float* W_fc  = (const float*)d_in[24];
    const float* b_fc  = (const float*)d_in[25];
    const float* W_res = (const float*)d_in[26];
    const float* b_res = (const float*)d_in[27];
    const float* W_time= (const float*)d_in[28];
    const float* b_time= (const float*)d_in[29];

    // workspace carve-up (floats)
    float* ws    = (float*)d_ws;
    float* eap   = ws;  ws += (size_t)N_EDGES * 4;
    float* xl    = ws;  ws += (size_t)N_NODES * CPROJ;
    float* xr    = ws;  ws += (size_t)N_NODES * CPROJ;
    float* exbuf = ws;  ws += (size_t)N_EDGES * NHEAD;   // score, then exp(in place)
    float* smax  = ws;  ws += (size_t)N_NODES * NHEAD;
    float* denom = ws;  ws += (size_t)N_NODES * NHEAD;
    float* agg   = ws;  ws += (size_t)N_NODES * CPROJ;
    float* hbuf  = ws;  ws += (size_t)N_NODES * HID;
    float* pool  = ws;  ws += (size_t)N_GRAPH * HID;
    float* cnt   = ws;  ws += (size_t)N_GRAPH;

    const int* src = ei;
    const int* dst = ei + N_EDGES;

    auto fill = [&](float* p, unsigned int v, long n) {
        fill_u32<<<(int)((n + 255) / 256), 256, 0, stream>>>((unsigned int*)p, v, n);
    };

    auto run_layer = [&](const float* xin, int K,
                         const float* Wl, const float* bl,
                         const float* Wr, const float* br,
                         const float* att, const float* We, const float* bias) {
        fill(smax,  0xFF800000u, (long)N_NODES * NHEAD);   // -inf
        fill(denom, 0u,          (long)N_NODES * NHEAD);
        fill(agg,   0u,          (long)N_NODES * CPROJ);

        dim3 ggrid(N_NODES / 16, CPROJ / 64);              // 3125 x 4
        gemm_wmma_f32<<<ggrid, 32, 0, stream>>>(xin, Wl, bl, xl, K);
        gemm_wmma_f32<<<ggrid, 32, 0, stream>>>(xin, Wr, br, xr, K);

        const long waves  = (long)N_EDGES * NHEAD;         // 1.6M (edge,head) waves
        const int wblocks = (int)((waves * 32 + 255) / 256);
        edge_scores<<<wblocks, 256, 0, stream>>>(xl, xr, eap, We, att, src, dst, exbuf, smax);
        edge_expsum<<<(int)((waves + 255) / 256), 256, 0, stream>>>(exbuf, dst, smax, exbuf, denom);
        edge_accum<<<wblocks, 256, 0, stream>>>(xl, src, dst, exbuf, denom, agg);
        node_finalize<<<(N_NODES * HID + 255) / 256, 256, 0, stream>>>(agg, bias, hbuf);
    };

    edge_embed<<<(N_EDGES + 255) / 256, 256, 0, stream>>>(
        ea, W_et, b_et, W_uw, b_uw, W_ua, b_ua, eap);

    run_layer(x,    IN_CH, Wl0, bl0, Wr0, br0, att0, We0, bias0);
    run_layer(hbuf, HID,   Wl1, bl1, Wr1, br1, att1, We1, bias1);

    fill(pool, 0u, (long)N_GRAPH * HID + N_GRAPH);         // pool + cnt contiguous
    pool_scatter<<<(N_NODES * HID + 255) / 256, 256, 0, stream>>>(hbuf, batch, pool, cnt);
    graph_head<<<N_GRAPH, HID, 0, stream>>>(pool, cnt, W_fc, b_fc,
                                            W_res, b_res, W_time, b_time,
                                            (float*)d_out);
}